// VerySimpleGCN_46179488367203
// MI455X (gfx1250) — compile-verified
//
#include <hip/hip_runtime.h>
#include <hip/hip_bf16.h>
#include <limits>

typedef __attribute__((ext_vector_type(2))) float v2f;
typedef __attribute__((ext_vector_type(8))) float v8f;

#define D 64

// ---------------------------------------------------------------------------
// Degree / dinv precompute (edge list is fixed across all 3 layers)
// ---------------------------------------------------------------------------
__global__ void init_deg_kernel(float* __restrict__ deg, int n) {
    int i = blockIdx.x * blockDim.x + threadIdx.x;
    if (i < n) deg[i] = 1.0f;   // self-loop contributes +1
}

__global__ void accum_deg_kernel(const int* __restrict__ dst, float* __restrict__ deg, int nE) {
    int e = blockIdx.x * blockDim.x + threadIdx.x;
    if (e < nE) atomicAdd(&deg[dst[e]], 1.0f);
}

__global__ void rsqrt_deg_kernel(float* __restrict__ deg, int n) {
    int i = blockIdx.x * blockDim.x + threadIdx.x;
    if (i < n) deg[i] = rsqrtf(deg[i]);   // in-place: deg -> dinv
}

// single-instruction ReLU: v_med3_f32(x, 0, +inf) == clamp to [0, inf)
__device__ __forceinline__ float relu1(float x) {
    return __builtin_amdgcn_fmed3f(x, 0.0f, __builtin_inff());
}

// ---------------------------------------------------------------------------
// WMMA GEMM: h[N,64] = act(in[N,64]) @ W[64,64]
// One wave -> 16x16 output tile via 16x V_WMMA_F32_16X16X4_F32 (K=64).
// Block = 256 threads = 8 waves = 32 rows x 64 cols.
// relu_in: apply ReLU to A-matrix elements on load (fuses prev layer's ReLU).
// ---------------------------------------------------------------------------
__global__ void gcn_gemm_wmma(const float* __restrict__ in, const float* __restrict__ W,
                              float* __restrict__ h, int n_rows, int relu_in) {
    const int wave = threadIdx.x >> 5;       // 0..7
    const int lane = threadIdx.x & 31;
    const int mrow = lane & 15;
    const int half = lane >> 4;              // 0 or 1
    const int khalf = half * 2;

    const int rowTile = blockIdx.x * 2 + (wave >> 2);
    const int colTile = wave & 3;
    const int row0 = rowTile * 16;
    const int col0 = colTile * 16;
    if (row0 >= n_rows) return;              // wave-uniform

    // clamp A row for safe loads on a (hypothetical) partial tile; EXEC stays full
    int rA = row0 + mrow;
    if (rA > n_rows - 1) rA = n_rows - 1;
    const float* arow = in + (size_t)rA * D;
    const float* wcol = W + col0 + mrow;

    v8f c = {0.f, 0.f, 0.f, 0.f, 0.f, 0.f, 0.f, 0.f};
#pragma unroll
    for (int k = 0; k < D; k += 4) {
        // A tile 16x4 f32: v0 = A[m][k+khalf], v1 = A[m][k+khalf+1]  (8B load)
        v2f a = *(const v2f*)(arow + k + khalf);
        if (relu_in) {
            a.x = relu1(a.x);
            a.y = relu1(a.y);
        }
        // B tile 4x16 f32 (mirrored layout): v0 = B[k+khalf][n], v1 = B[k+khalf+1][n]
        v2f b;
        b.x = wcol[(k + khalf) * D];
        b.y = wcol[(k + khalf + 1) * D];
        c = __builtin_amdgcn_wmma_f32_16x16x4_f32(
                /*neg_a=*/false, a, /*neg_b=*/false, b,
                /*c_mod=*/(short)0, c, /*reuse_a=*/false, /*reuse_b=*/false);
    }

    // D layout: VGPR i -> row (row0 + i + half*8), col (col0 + mrow)
#pragma unroll
    for (int i = 0; i < 8; ++i) {
        int r = row0 + i + half * 8;
        if (r < n_rows) h[(size_t)r * D + col0 + mrow] = c[i];
    }
}

// ---------------------------------------------------------------------------
// agg[i][f] = h[i][f] * dinv[i]^2 + bias[f]   (self-loop term + bias init)
// One float4 per thread: b128 in / b128 out.
// ---------------------------------------------------------------------------
__global__ void gcn_selfloop_bias(const float* __restrict__ h, const float* __restrict__ dinv,
                                  const float* __restrict__ bias, float* __restrict__ agg,
                                  int n_nodes) {
    int t = blockIdx.x * blockDim.x + threadIdx.x;   // float4 index
    int total = n_nodes * (D / 4);
    if (t >= total) return;
    int node = t >> 4;                               // 16 float4 per node
    int q = t & 15;
    float di = dinv[node];
    float s = di * di;
    float4 v = ((const float4*)h)[t];
    float4 bi = ((const float4*)bias)[q];
    float4 r;
    r.x = v.x * s + bi.x;
    r.y = v.y * s + bi.y;
    r.z = v.z * s + bi.z;
    r.w = v.w * s + bi.w;
    ((float4*)agg)[t] = r;
}

// ---------------------------------------------------------------------------
// Edge scatter-add: 16 lanes per edge, one float4 per lane (coalesced 256B
// gather of h[src]); fire-and-forget f32 atomics into agg[dst] (L2-resident).
// ---------------------------------------------------------------------------
__global__ void gcn_edge_agg(const int* __restrict__ src, const int* __restrict__ dst,
                             const float* __restrict__ h, const float* __restrict__ dinv,
                             float* __restrict__ agg, int n_edges) {
    int t = blockIdx.x * blockDim.x + threadIdx.x;
    int e = t >> 4;
    if (e >= n_edges) return;
    int f4 = (t & 15) * 4;
    int s = src[e];
    int d = dst[e];
    float norm = dinv[s] * dinv[d];
    const float4* hp = (const float4*)(h + (size_t)s * D + f4);
    float4 v = *hp;
    float* ap = agg + (size_t)d * D + f4;
    atomicAdd(ap + 0, v.x * norm);
    atomicAdd(ap + 1, v.y * norm);
    atomicAdd(ap + 2, v.z * norm);
    atomicAdd(ap + 3, v.w * norm);
}

// ---------------------------------------------------------------------------
// Launch
// ---------------------------------------------------------------------------
extern "C" void kernel_launch(void* const* d_in, const int* in_sizes, int n_in,
                              void* d_out, int out_size, void* d_ws, size_t ws_size,
                              hipStream_t stream) {
    const float* x  = (const float*)d_in[0];
    const int*   ei = (const int*)d_in[1];
    const float* W0 = (const float*)d_in[2];
    const float* b0 = (const float*)d_in[3];
    const float* W1 = (const float*)d_in[4];
    const float* b1 = (const float*)d_in[5];
    const float* W2 = (const float*)d_in[6];
    const float* b2 = (const float*)d_in[7];
    float* out = (float*)d_out;

    const int N = in_sizes[0] / D;          // 100000
    const int E = in_sizes[1] / 2;          // 1600000
    const int* src = ei;
    const int* dst = ei + E;

    // workspace layout (floats): dinv[N] | h[N*64] | agg[N*64]
    float* ws   = (float*)d_ws;
    size_t nPad = ((size_t)N + 63) & ~(size_t)63;
    float* dinv = ws;
    float* h    = ws + nPad;
    float* agg  = h + (size_t)N * D;

    const int TB = 256;
    dim3 blk(TB);

    // --- degree / dinv (once; edge list fixed) ---
    init_deg_kernel<<<dim3((N + TB - 1) / TB), blk, 0, stream>>>(dinv, N);
    accum_deg_kernel<<<dim3((E + TB - 1) / TB), blk, 0, stream>>>(dst, dinv, E);
    rsqrt_deg_kernel<<<dim3((N + TB - 1) / TB), blk, 0, stream>>>(dinv, N);

    dim3 gemmGrid((N + 31) / 32);                          // 32 rows / block
    dim3 mapGrid(((size_t)N * (D / 4) + TB - 1) / TB);     // one float4 / thread
    dim3 edgeGrid(((size_t)E * 16 + TB - 1) / TB);

    // --- layer 0: h = x@W0 ; agg = h*dinv^2 + b0 ; += edge messages ---
    gcn_gemm_wmma<<<gemmGrid, blk, 0, stream>>>(x, W0, h, N, 0);
    gcn_selfloop_bias<<<mapGrid, blk, 0, stream>>>(h, dinv, b0, agg, N);
    gcn_edge_agg<<<edgeGrid, blk, 0, stream>>>(src, dst, h, dinv, agg, E);

    // --- layer 1: ReLU fused into GEMM A-load ---
    gcn_gemm_wmma<<<gemmGrid, blk, 0, stream>>>(agg, W1, h, N, 1);
    gcn_selfloop_bias<<<mapGrid, blk, 0, stream>>>(h, dinv, b1, agg, N);
    gcn_edge_agg<<<edgeGrid, blk, 0, stream>>>(src, dst, h, dinv, agg, E);

    // --- layer 2: ReLU fused; final agg written straight to d_out (no ReLU) ---
    gcn_gemm_wmma<<<gemmGrid, blk, 0, stream>>>(agg, W2, h, N, 1);
    gcn_selfloop_bias<<<mapGrid, blk, 0, stream>>>(h, dinv, b2, out, N);
    gcn_edge_agg<<<edgeGrid, blk, 0, stream>>>(src, dst, h, dinv, out, E);
}